// RBFPolicyNetwork_22213570855524
// MI455X (gfx1250) — compile-verified
//
#include <hip/hip_runtime.h>
#include <math.h>

typedef float v2f __attribute__((ext_vector_type(2)));
typedef float v8f __attribute__((ext_vector_type(8)));

#define NB 256   // num basis
#define NO 32    // output size
#define NI 128   // input size

// One workgroup per (basis, output) pair: streams the 128x128 f32 Lambda_inv
// slice once (64 KB), computes quad = d^T M d via v_wmma_f32_16x16x4_f32,
// writes rbf = exp(-0.5*quad) to workspace.
__global__ __launch_bounds__(128) void rbf_quad_kernel(
    const float* __restrict__ x,
    const float* __restrict__ centers,
    const float* __restrict__ Lambda_inv,
    float* __restrict__ rbf_out)
{
    __shared__ float dloc[NI];
    __shared__ float partial[4];

    const int pair  = blockIdx.x;      // pair = b*32 + o
    const int tid   = threadIdx.x;     // 0..127
    const int lane  = tid & 31;        // wave32 lane
    const int wave  = tid >> 5;        // 0..3

    // diff = x - centers[b][o][:]  (centers is (B,O,I) row-major -> pair*128)
    dloc[tid] = x[tid] - centers[(size_t)pair * NI + tid];
    __syncthreads();

    const float* __restrict__ M = Lambda_inv + (size_t)pair * NI * NI;

    const int lrow  = lane & 15;   // A-matrix M index (row)
    const int lhalf = lane >> 4;   // 0: K=0,1  1: K=2,3  (ISA 16x4 f32 A layout)

    float part = 0.0f;

    // Each wave owns two 16-row strips of M: rows [16r, 16r+16)
    for (int s = 0; s < 2; ++s) {
        const int r = wave * 2 + s;                     // strip 0..7
        const float* rowp = M + (size_t)(16 * r + lrow) * NI + 2 * lhalf;
        v8f acc = {};
#pragma unroll 8
        for (int kt = 0; kt < NI / 4; ++kt) {
            if ((kt & 7) == 0) {
                // stream-ahead prefetch (~1KB ahead) -> global_prefetch_b8
                __builtin_prefetch(rowp + 4 * kt + 256, 0, 1);
            }
            // A fragment: A[m][k] = M[16r+m][4kt+k]; lane holds k = 2*lhalf + {0,1}
            v2f a = *(const v2f*)(rowp + 4 * kt);
            // B fragment: B[k][n] = d[4kt+k] broadcast over all n columns.
            // Layout (by symmetry with C/D half-split): vgpr g holds K = g + 2*(lane/16)
            v2f bf;
            bf.x = dloc[4 * kt + 2 * lhalf];
            bf.y = dloc[4 * kt + 2 * lhalf + 1];
            acc = __builtin_amdgcn_wmma_f32_16x16x4_f32(
                false, a, false, bf, (short)0, acc, false, false);
        }
        // D layout: vgpr g, lane l -> D[8*(l/16)+g][l%16]; all 16 columns equal y[m].
        // Local dot with u[m] = d[16r + m]; wave sum yields 16x strip contribution.
        const float* u = &dloc[16 * r + 8 * lhalf];
#pragma unroll
        for (int g = 0; g < 8; ++g) part += u[g] * acc[g];
    }

    // wave32 reduction
    for (int off = 16; off > 0; off >>= 1)
        part += __shfl_xor(part, off, 32);
    if (lane == 0) partial[wave] = part;
    __syncthreads();

    if (tid == 0) {
        float quad = (partial[0] + partial[1] + partial[2] + partial[3]) * (1.0f / 16.0f);
        rbf_out[pair] = __expf(-0.5f * quad);
    }
}

// Tiny epilogue: policy[o] = tanh(sum_b w[o,b]*rbf[b,o]); rescale to bounds.
__global__ __launch_bounds__(64) void policy_kernel(
    const float* __restrict__ rbf,      // (B, O) = 8192 floats in d_ws
    const float* __restrict__ weights,  // (O, B)
    const float* __restrict__ bnds,     // (2, O)
    float* __restrict__ out)            // (O)
{
    int o = threadIdx.x;
    if (o < NO) {
        float s = 0.0f;
#pragma unroll 8
        for (int b = 0; b < NB; ++b)
            s += weights[o * NB + b] * rbf[(size_t)b * NO + o];
        float p  = tanhf(s);
        float lo = bnds[o];
        float hi = bnds[NO + o];
        out[o] = (p + 1.0f) * 0.5f * (hi - lo) + lo;
    }
}

extern "C" void kernel_launch(void* const* d_in, const int* in_sizes, int n_in,
                              void* d_out, int out_size, void* d_ws, size_t ws_size,
                              hipStream_t stream) {
    const float* x       = (const float*)d_in[0];   // (128)
    const float* weights = (const float*)d_in[1];   // (32, 256)
    const float* centers = (const float*)d_in[2];   // (256, 32, 128)
    const float* Lambda  = (const float*)d_in[3];   // (256, 32, 128, 128)
    const float* bnds    = (const float*)d_in[4];   // (2, 32)
    float* rbf = (float*)d_ws;                      // 8192 floats scratch
    float* out = (float*)d_out;                     // (32)

    rbf_quad_kernel<<<NB * NO, 128, 0, stream>>>(x, centers, Lambda, rbf);
    policy_kernel<<<1, 64, 0, stream>>>(rbf, weights, bnds, out);
}